// Layer_70411693850653
// MI455X (gfx1250) — compile-verified
//
#include <hip/hip_runtime.h>
#include <hip/hip_bf16.h>

typedef float v2f __attribute__((ext_vector_type(2)));
typedef float v8f __attribute__((ext_vector_type(8)));

#define MUL 32
#define NF_COLS 128
#define SUM_COLS 384   // 4 channels x 96 contraction entries (tp1e/tp2e are dead code)
#define K_DIM 96
#define V_DIM 32
#define XR_STRIDE 100  // padded LDS row stride (floats): 16B aligned, bank-conflict-free

__device__ __forceinline__ void atomAddF32(float* p, float v) {
    // no-return global f32 atomic add (STOREcnt)
    __hip_atomic_fetch_add(p, v, __ATOMIC_RELAXED, __HIP_MEMORY_SCOPE_AGENT);
}

// ---------------- zero workspace (vectorized) ----------------
__global__ void zero_kernel(float* __restrict__ p, size_t n) {
    size_t n4 = n >> 2;
    float4* p4 = (float4*)p;
    size_t i = (size_t)blockIdx.x * blockDim.x + threadIdx.x;
    size_t stride = (size_t)gridDim.x * blockDim.x;
    float4 z = make_float4(0.f, 0.f, 0.f, 0.f);
    for (size_t j = i; j < n4; j += stride) p4[j] = z;
    for (size_t j = (n4 << 2) + i; j < n; j += stride) p[j] = 0.0f;
}

// ---------------- fold skip-connection + scalings into effective weights ----------------
// Weff[k,v] = W[k,v]*inv96/1.5 + (k<32 ? Ws[k,v]*inv32 : 0)
// Stored PAIRED for the GEMM: Wp[(k/2)*64 + 2*v + (k&1)]  (one b64 load per lane/iter)
__global__ void weff_kernel(const float* __restrict__ W0, const float* __restrict__ W1,
                            const float* __restrict__ Ws0, const float* __restrict__ Ws1,
                            float* __restrict__ W0p, float* __restrict__ W1p) {
    const float inv96d = 0.10206207261596575f * (1.0f / 1.5f); // 1/sqrt(96)/1.5
    const float inv32  = 0.17677669529663687f;                 // 1/sqrt(32)
    int idx = blockIdx.x * blockDim.x + threadIdx.x;
    if (idx >= 2 * K_DIM * V_DIM) return;
    int which = idx >= K_DIM * V_DIM;
    int kv = idx - which * K_DIM * V_DIM;
    int k = kv / V_DIM;
    int v = kv - k * V_DIM;
    int pos = (k >> 1) * (2 * V_DIM) + 2 * v + (k & 1);
    if (!which) {
        float w = W0[kv] * inv96d;
        if (k < MUL) w += Ws0[kv] * inv32;
        W0p[pos] = w;
    } else {
        float w = W1[kv] * inv96d;
        if (k < MUL) w += Ws1[kv] * inv32;
        W1p[pos] = w;
    }
}

// ---------------- per-edge gather + tensor-product + scatter-add ----------------
// One thread per (edge, u). Scatters DIRECTLY into the channel-major layout
// sums[node][ch][k]: ch0 = {f0, p00, dot}, ch c+1 = {f1_c, p01_c, p10_c}.
// All 12 atomics are at (dst + u) + constant offsets.
__global__ __launch_bounds__(256) void edge_kernel(
    const float* __restrict__ nfeat, const float* __restrict__ sh,
    const int* __restrict__ senders, const int* __restrict__ receivers,
    float* __restrict__ sums, float* __restrict__ cnt, int E) {
    const float INV_SQRT3 = 0.5773502691896258f;
    int e = blockIdx.x * 8 + (threadIdx.x >> 5);
    int u = threadIdx.x & 31;
    if (e >= E) return;

    int s = senders[e];
    int r = receivers[e];
    float4 shv = ((const float4*)sh)[e];
    float s0 = shv.x, ax = shv.y, ay = shv.z, az = shv.w;

    const float* nf = nfeat + (size_t)s * NF_COLS;
    float f0u = nf[u];
    int c1 = 32 + 3 * u;
    float bx = nf[c1 + 0];
    float by = nf[c1 + 1];
    float bz = nf[c1 + 2];

    float* dst = sums + (size_t)r * SUM_COLS + u;
    // ch0: f0 | p00 | dot
    atomAddF32(dst + 0,  f0u);
    atomAddF32(dst + 32, s0 * f0u);
    atomAddF32(dst + 64, (ax * bx + ay * by + az * bz) * INV_SQRT3);
    // ch1 (x): f1x | p01x | p10x
    atomAddF32(dst + 96,  bx);
    atomAddF32(dst + 128, s0 * bx);
    atomAddF32(dst + 160, f0u * ax);
    // ch2 (y)
    atomAddF32(dst + 192, by);
    atomAddF32(dst + 224, s0 * by);
    atomAddF32(dst + 256, f0u * ay);
    // ch3 (z)
    atomAddF32(dst + 288, bz);
    atomAddF32(dst + 320, s0 * bz);
    atomAddF32(dst + 352, f0u * az);
    if (u == 0) atomAddF32(cnt + r, 1.0f);
}

// ---------------- WMMA output GEMM ----------------
// Grid: N/16 blocks of 256 threads (8 waves). Staging is a verbatim scaled copy
// (sums already channel-major). LDS rows are [ch*16+node] with stride 100 floats
// so the per-lane ds_load_2addr_b64 A-loads are bank-conflict-free.
// Hot loop per wave: ds_load_b64 (imm offset) + global_load_b64 (imm) + wmma.
__global__ __launch_bounds__(256) void gemm_kernel(
    const float* __restrict__ sums, const float* __restrict__ cnt,
    const float* __restrict__ W0p, const float* __restrict__ W1p,
    float* __restrict__ out, int N) {
    __shared__ float Xr[64 * XR_STRIDE];  // 25.6 KB, row = ch*16 + node
    __shared__ float Os[16 * NF_COLS];    // 8 KB output staging
    __shared__ float cinvs[16];

    int tid = threadIdx.x;
    int base = blockIdx.x * 16;           // N is a multiple of 16

    if (tid < 16) cinvs[tid] = 1.0f / fmaxf(cnt[base + tid], 1.0f);
    __syncthreads();

    // ---- stage + prescale (coalesced b128 reads, b128 LDS stores) ----
    {
        const float4* src = (const float4*)(sums + (size_t)base * SUM_COLS);
#pragma unroll
        for (int i = 0; i < 6; ++i) {
            int flat4 = tid + i * 256;            // 0..1535
            int rsrc = flat4 / 24;                // node*4 + ch   (24 float4s per row)
            int k4 = flat4 - rsrc * 24;
            int row = ((rsrc & 3) << 4) + (rsrc >> 2);  // -> ch*16 + node
            float4 v = src[flat4];
            float ci = cinvs[rsrc >> 2];
            v.x *= ci; v.y *= ci; v.z *= ci; v.w *= ci;
            *(float4*)&Xr[row * XR_STRIDE + (k4 << 2)] = v;
        }
    }
    __syncthreads();

    // ---- WMMA: wave w -> channel ch=w>>1, output half vtile=w&1 ----
    int lane = tid & 31;
    int wave = tid >> 5;
    int ch = wave >> 1;          // 0..3
    int vtile = wave & 1;        // 0..1
    int lm = lane & 15;
    int half = lane >> 4;
    int kb = 2 * half;           // per-lane K sub-offset

    int bcol = vtile * 16 + lm;
    const float* xp = &Xr[(ch * 16 + lm) * XR_STRIDE + kb];
    const v2f* bpv = (const v2f*)((ch == 0) ? W0p : W1p);
    int bbase = half * V_DIM + bcol;                     // (kb/2)*32 + bcol

    v8f acc = {};
#pragma unroll
    for (int k0 = 0; k0 < K_DIM; k0 += 4) {
        v2f a = *(const v2f*)(xp + k0);                  // ds_load_b64, imm offset
        v2f b = bpv[bbase + (k0 >> 1) * V_DIM];          // global_load_b64, imm offset
        acc = __builtin_amdgcn_wmma_f32_16x16x4_f32(
            false, a, false, b, (short)0, acc, false, false);
    }

    // ---- D -> LDS staging, then coalesced float4 write-out ----
    int outcol = (ch == 0) ? bcol : (32 + 3 * bcol + ch - 1);
#pragma unroll
    for (int r = 0; r < 8; ++r)
        Os[(r + 8 * half) * NF_COLS + outcol] = acc[r];
    __syncthreads();

    {
        const float4* s4 = (const float4*)Os;
        float4* o4 = (float4*)(out + (size_t)base * NF_COLS);
#pragma unroll
        for (int i = 0; i < (16 * NF_COLS / 4) / 256; ++i)
            o4[tid + i * 256] = s4[tid + i * 256];
    }
}

extern "C" void kernel_launch(void* const* d_in, const int* in_sizes, int n_in,
                              void* d_out, int out_size, void* d_ws, size_t ws_size,
                              hipStream_t stream) {
    const float* nfeat = (const float*)d_in[0];
    const float* sh    = (const float*)d_in[1];
    const int* senders = (const int*)d_in[2];
    const int* recv    = (const int*)d_in[3];
    const float* W0    = (const float*)d_in[4];
    const float* W1    = (const float*)d_in[5];
    const float* Ws0   = (const float*)d_in[6];
    const float* Ws1   = (const float*)d_in[7];
    float* out = (float*)d_out;

    int N = in_sizes[0] / NF_COLS;   // 50000
    int E = in_sizes[2];             // 200000

    float* ws = (float*)d_ws;
    float* sums = ws;                               // N*384 (channel-major per node)
    float* cnt  = sums + (size_t)N * SUM_COLS;      // N
    float* W0p  = cnt + N;                          // 96*32 (paired layout)
    float* W1p  = W0p + K_DIM * V_DIM;              // 96*32 (paired layout)

    size_t nzero = (size_t)N * (SUM_COLS + 1);
    zero_kernel<<<2048, 256, 0, stream>>>(ws, nzero);

    weff_kernel<<<(2 * K_DIM * V_DIM + 255) / 256, 256, 0, stream>>>(
        W0, W1, Ws0, Ws1, W0p, W1p);

    edge_kernel<<<(E + 7) / 8, 256, 0, stream>>>(
        nfeat, sh, senders, recv, sums, cnt, E);

    gemm_kernel<<<N / 16, 256, 0, stream>>>(
        sums, cnt, W0p, W1p, out, N);
}